// PowerWhitening_86354612453444
// MI455X (gfx1250) — compile-verified
//
#include <hip/hip_runtime.h>
#include <hip/hip_bf16.h>

#define D       128
#define N_ROWS  524288
#define N_PITER 100

typedef __attribute__((ext_vector_type(2))) float v2f;
typedef __attribute__((ext_vector_type(8))) float v8f;
typedef __attribute__((ext_vector_type(4))) int   v4i;

// workspace layout (floats)
#define GRAM_OFF   0                    // 128*128
#define COLSUM_OFF 16384                // 128
#define CMAT_OFF   16512                // 128*128
#define WMAT_OFF   (16512 + 16384)      // 128*128
#define MEAN_OFF   (WMAT_OFF + 16384)   // 128

// ---------------------------------------------------------------------------
// CDNA5 async global->LDS helpers (ASYNCcnt path), with safe fallbacks.
// Builtin signature (from hipcc diagnostic): first arg is v4i in addrspace(1),
// i.e. (global v4i*, lds v4i*, imm offset, imm cpol).
// ---------------------------------------------------------------------------
__device__ __forceinline__ void async_copy_b128(const float* gsrc, float* ldst) {
#if defined(__has_builtin) && __has_builtin(__builtin_amdgcn_global_load_async_to_lds_b128)
    void* g = (void*)gsrc;   // strip const
    void* l = (void*)ldst;
    __builtin_amdgcn_global_load_async_to_lds_b128(
        (__attribute__((address_space(1))) v4i*)g,
        (__attribute__((address_space(3))) v4i*)l,
        /*offset=*/0, /*cpol=*/0);
#else
    *(float4*)ldst = *(const float4*)gsrc;   // synchronous fallback
#endif
}

template <int CNT>
__device__ __forceinline__ void wait_asynccnt() {
#if defined(__has_builtin) && __has_builtin(__builtin_amdgcn_s_wait_asynccnt)
    __builtin_amdgcn_s_wait_asynccnt(CNT);
#else
    asm volatile("s_wait_asynccnt %0" ::"n"(CNT) : "memory");
#endif
}

// ---------------------------------------------------------------------------
// Kernel 0: zero Gram + colsum accumulators (ws is poisoned by the harness)
// ---------------------------------------------------------------------------
__global__ void pw_zero_kernel(float* __restrict__ ws) {
    int idx = blockIdx.x * blockDim.x + threadIdx.x;
    if (idx < 16512) ws[idx] = 0.0f;
}

// ---------------------------------------------------------------------------
// Kernel 1: split-K Gram = x^T x  (f32 WMMA 16x16x4)  + column sums.
// 256 WGs x 256 threads (8 wave32). 16-row x tiles are DMA'd straight into
// LDS with GLOBAL_LOAD_ASYNC_TO_LDS_B128 (double buffered: fill stage s+1
// while 32 WMMAs consume stage s). Per-wave ASYNCcnt never exceeds 4.
// ---------------------------------------------------------------------------
__global__ void pw_gram_kernel(const float* __restrict__ x,
                               float* __restrict__ gram,
                               float* __restrict__ colsum) {
    __shared__ float xs[2][16 * D];   // 2 x 8 KB staging tiles
    const int tid   = threadIdx.x;
    const int wave  = tid >> 5;
    const int lane  = tid & 31;
    const int laneM = lane & 15;
    const bool hi   = (lane >= 16);

    const int chunk   = N_ROWS / gridDim.x;     // 2048
    const int rowBase = blockIdx.x * chunk;
    const int stages  = chunk >> 4;             // 128 stages of 16 rows

    v8f acc[8];
#pragma unroll
    for (int n = 0; n < 8; ++n)
#pragma unroll
        for (int r = 0; r < 8; ++r) acc[n][r] = 0.0f;
    float csum = 0.0f;

    // prologue: kick off stage 0 fill (2 x b128 per thread = 512 float4)
    {
        const float* src = x + (size_t)rowBase * D;
        async_copy_b128(src + tid * 4,         &xs[0][tid * 4]);
        async_copy_b128(src + (tid + 256) * 4, &xs[0][(tid + 256) * 4]);
    }

    for (int s = 0; s < stages; ++s) {
        const int buf = s & 1;
        if (s + 1 < stages) {   // fill next stage while this one computes
            const float* src = x + (size_t)(rowBase + ((s + 1) << 4)) * D;
            async_copy_b128(src + tid * 4,         &xs[buf ^ 1][tid * 4]);
            async_copy_b128(src + (tid + 256) * 4, &xs[buf ^ 1][(tid + 256) * 4]);
            wait_asynccnt<2>();     // retire stage s (issued before these 2)
        } else {
            wait_asynccnt<0>();
        }
        __syncthreads();

        if (tid < D) {                      // column-sum partial for mean
#pragma unroll
            for (int r = 0; r < 16; ++r) csum += xs[buf][r * D + tid];
        }

#pragma unroll
        for (int k4 = 0; k4 < 4; ++k4) {    // 4 K-substeps of depth 4
            const int k = k4 << 2;
            // A fragment: A[M][K] = x[k+K][16*wave+M]   (x^T tile)
            v2f a;
            const int am = (wave << 4) + laneM;
            a.x = xs[buf][(k + (hi ? 2 : 0)) * D + am];
            a.y = xs[buf][(k + (hi ? 3 : 1)) * D + am];
#pragma unroll
            for (int n = 0; n < 8; ++n) {
                // B fragment: B[K][Ncol] = x[k+K][16*n+Ncol]
                v2f b;
                const int bn = (n << 4) + laneM;
                b.x = xs[buf][(k + (hi ? 1 : 0)) * D + bn];
                b.y = xs[buf][(k + (hi ? 3 : 2)) * D + bn];
                acc[n] = __builtin_amdgcn_wmma_f32_16x16x4_f32(
                    false, a, false, b, (short)0, acc[n], false, false);
            }
        }
        __syncthreads();   // protect buffer reuse by the s+2 fill
    }

    if (tid < D) atomicAdd(&colsum[tid], csum);

    const int gRow = (wave << 4) + (hi ? 8 : 0);
#pragma unroll
    for (int n = 0; n < 8; ++n)
#pragma unroll
        for (int r = 0; r < 8; ++r)
            atomicAdd(&gram[(gRow + r) * D + (n << 4) + laneM], acc[n][r]);
}

// ---------------------------------------------------------------------------
// Kernel 2: C0 = Gram/N - mean*mean^T ; also emit mean[]
// ---------------------------------------------------------------------------
__global__ void pw_finalize_kernel(const float* __restrict__ gram,
                                   const float* __restrict__ colsum,
                                   float* __restrict__ Cmat,
                                   float* __restrict__ mean) {
    const int idx = blockIdx.x * blockDim.x + threadIdx.x;
    const float invN = 1.0f / (float)N_ROWS;
    if (idx < D) mean[idx] = colsum[idx] * invN;
    if (idx < D * D) {
        const int i = idx >> 7, j = idx & (D - 1);
        Cmat[idx] = gram[idx] * invN - (colsum[i] * invN) * (colsum[j] * invN);
    }
}

// ---------------------------------------------------------------------------
// Kernel 3: sequential power iteration + deflation. One workgroup; C lives in
// LDS (64 KB of the 320 KB WGP pool). 256 threads: 2 threads per matrix row,
// wave32 shuffle reductions, 3 barriers per inner iteration.
// ---------------------------------------------------------------------------
__global__ void pw_power_kernel(const float* __restrict__ R,
                                const float* __restrict__ Cmat,
                                float* __restrict__ W) {
    extern __shared__ float smem[];
    float* Cs  = smem;            // 16384
    float* vsh = Cs + D * D;      // 128
    float* red = vsh + D;         // 8
    float* sc  = red + 8;         // 1

    const int tid  = threadIdx.x;
    const int wave = tid >> 5;
    const int lane = tid & 31;
    const int row  = tid >> 1;
    const int half = tid & 1;

    for (int i = tid; i < D * D; i += 256) { Cs[i] = Cmat[i]; W[i] = 0.0f; }
    __syncthreads();

    for (int ev = 0; ev < D; ++ev) {
        if (tid < D) vsh[tid] = R[tid * D + ev];   // v0 = R[:, ev]
        __syncthreads();

        float e = 0.0f;
        for (int it = 0; it <= N_PITER; ++it) {
            // w[row] = C[row,:] . v   (2 threads per row, 64 MACs each)
            float s = 0.0f;
            const float* crow = Cs + row * D + half * 64;
            const float* vv   = vsh + half * 64;
#pragma unroll 8
            for (int j = 0; j < 64; ++j) s += crow[j] * vv[j];
            s += __shfl_xor(s, 1);                 // combine row halves

            float q = half ? 0.0f : s * s;         // ||w||^2 partials
#pragma unroll
            for (int off = 16; off > 0; off >>= 1) q += __shfl_xor(q, off);
            if (lane == 0) red[wave] = q;
            __syncthreads();
            if (tid == 0) {
                float n2 = 0.0f;
                for (int k = 0; k < 8; ++k) n2 += red[k];
                sc[0] = n2;
            }
            __syncthreads();
            const float n2 = sc[0];
            if (it == N_PITER) { e = sqrtf(n2); break; }  // e = ||C v||
            if (half == 0) vsh[row] = s * rsqrtf(n2);     // v = w/||w||
            __syncthreads();
        }

        // deflate C and accumulate whitening matrix: W += v v^T / sqrt(e)
        const float rse = rsqrtf(e);
        for (int idx = tid; idx < D * D; idx += 256) {
            const int i = idx >> 7, j = idx & (D - 1);
            const float vv = vsh[i] * vsh[j];
            Cs[idx] -= e * vv;
            W[idx]  += vv * rse;
        }
        __syncthreads();
    }
}

// ---------------------------------------------------------------------------
// Kernel 4: out = (x - mean) @ W^T  (f32 WMMA). 4096 WGs x 128 rows.
// W^T staged once per WG in LDS (64 KB); A fragments streamed from global.
// ---------------------------------------------------------------------------
__global__ void pw_out_kernel(const float* __restrict__ x,
                              const float* __restrict__ W,
                              const float* __restrict__ mean,
                              float* __restrict__ out) {
    extern __shared__ float smem[];
    float* Wt = smem;                          // Wt[j][c] = W[c][j], 64 KB
    const int tid   = threadIdx.x;
    const int wave  = tid >> 5;
    const int lane  = tid & 31;
    const int laneM = lane & 15;
    const bool hi   = (lane >= 16);

    for (int idx = tid; idx < D * D; idx += 256) {
        const int c = idx >> 7, j = idx & (D - 1);
        Wt[j * D + c] = W[idx];
    }
    __syncthreads();

    const int rowBase = blockIdx.x * 128 + (wave << 4);
    v8f acc[8];
#pragma unroll
    for (int n = 0; n < 8; ++n)
#pragma unroll
        for (int r = 0; r < 8; ++r) acc[n][r] = 0.0f;

    for (int k = 0; k < D; k += 4) {
        // A fragment: xc[rowBase+M][k+K]
        const int m  = rowBase + laneM;
        const int k0 = k + (hi ? 2 : 0);
        const float2 av = *(const float2*)(x + (size_t)m * D + k0);
        v2f a;
        a.x = av.x - mean[k0];
        a.y = av.y - mean[k0 + 1];
#pragma unroll
        for (int n = 0; n < 8; ++n) {
            // B fragment: B[K][Ncol] = Wt[k+K][16*n+Ncol]
            v2f b;
            const int bn = (n << 4) + laneM;
            b.x = Wt[(k + (hi ? 1 : 0)) * D + bn];
            b.y = Wt[(k + (hi ? 3 : 2)) * D + bn];
            acc[n] = __builtin_amdgcn_wmma_f32_16x16x4_f32(
                false, a, false, b, (short)0, acc[n], false, false);
        }
    }

    const int mBase = rowBase + (hi ? 8 : 0);
#pragma unroll
    for (int n = 0; n < 8; ++n)
#pragma unroll
        for (int r = 0; r < 8; ++r)
            out[(size_t)(mBase + r) * D + (n << 4) + laneM] = acc[n][r];
}

// ---------------------------------------------------------------------------
extern "C" void kernel_launch(void* const* d_in, const int* in_sizes, int n_in,
                              void* d_out, int out_size, void* d_ws, size_t ws_size,
                              hipStream_t stream) {
    const float* x = (const float*)d_in[0];
    const float* R = (const float*)d_in[1];
    float* out = (float*)d_out;
    float* ws  = (float*)d_ws;

    float* gram   = ws + GRAM_OFF;
    float* colsum = ws + COLSUM_OFF;
    float* Cmat   = ws + CMAT_OFF;
    float* Wmat   = ws + WMAT_OFF;
    float* mean   = ws + MEAN_OFF;

    pw_zero_kernel<<<65, 256, 0, stream>>>(ws);
    pw_gram_kernel<<<256, 256, 0, stream>>>(x, gram, colsum);
    pw_finalize_kernel<<<64, 256, 0, stream>>>(gram, colsum, Cmat, mean);
    pw_power_kernel<<<1, 256, (D * D + D + 16) * sizeof(float), stream>>>(R, Cmat, Wmat);
    pw_out_kernel<<<N_ROWS / 128, 256, D * D * sizeof(float), stream>>>(x, Wmat, mean, out);
}